// InputLayer_48670569398535
// MI455X (gfx1250) — compile-verified
//
#include <hip/hip_runtime.h>
#include <hip/hip_bf16.h>

// ---------------------------------------------------------------------------
// out[i,j,k] = LN_k( (x[i,j]*w[k]+b[k]) * sqrt(512) ) * gamma[k] + beta[k]
// Collapsed analytically to: out = (x*inv)*P_k + inv*Q_k + 1*beta_k
//   P_k = s*(w_k - mw)*gamma_k,  Q_k = s*(b_k - mb)*gamma_k
//   inv = rsqrt( s^2*(x^2*E[dw^2] + 2x*E[dw*db] + E[db^2]) + eps )
// Mapped onto V_WMMA_F32_16X16X4_F32 using all of K=4:
//   K0 = x*inv row,  K1 = inv row,  K2 = ones row (carries beta),  K3 = 0
// ---------------------------------------------------------------------------

#define SIZE 512
#define EPSV 1e-5f
#define SCALE_S 22.62741699796952f /* sqrt(512) */

typedef float v2f __attribute__((ext_vector_type(2)));
typedef float v8f __attribute__((ext_vector_type(8)));

// ---------------- Stage 1: per-channel coefficient precompute ----------------
// ws layout (floats):
//   [0 .. 1023]    packed {P_k, Q_k} pairs        (512 x v2f)
//   [1024 .. 2047] packed {beta_k, 0} pairs       (512 x v2f)
//   [2048..2050]   quadratic coefs a, b2, c(+eps)

__device__ inline float block_reduce_sum_512(float v, float* red) {
    int k = threadIdx.x;
    red[k] = v;
    __syncthreads();
#pragma unroll
    for (int s = 256; s >= 1; s >>= 1) {
        if (k < s) red[k] += red[k + s];
        __syncthreads();
    }
    float r = red[0];
    __syncthreads();  // safe LDS reuse
    return r;
}

__global__ void ln_stats_kernel(const float* __restrict__ w,
                                const float* __restrict__ b,
                                const float* __restrict__ gamma,
                                const float* __restrict__ beta,
                                float* __restrict__ ws) {
    __shared__ float red[SIZE];
    const int k = threadIdx.x;
    const float wv = w[k];
    const float bv = b[k];
    const float gv = gamma[k];

    const float mw = block_reduce_sum_512(wv, red) * (1.0f / SIZE);
    const float mb = block_reduce_sum_512(bv, red) * (1.0f / SIZE);

    const float dw = wv - mw;
    const float db = bv - mb;

    const float sdw2 = block_reduce_sum_512(dw * dw, red) * (1.0f / SIZE);
    const float sdwb = block_reduce_sum_512(dw * db, red) * (1.0f / SIZE);
    const float sdb2 = block_reduce_sum_512(db * db, red) * (1.0f / SIZE);

    ws[2 * k + 0]        = SCALE_S * dw * gv;   // P_k
    ws[2 * k + 1]        = SCALE_S * db * gv;   // Q_k
    ws[1024 + 2 * k + 0] = beta[k];             // beta row (K=2)
    ws[1024 + 2 * k + 1] = 0.0f;                // K=3 row
    if (k == 0) {
        // var + eps = x*(x*a + b2) + c
        ws[2048] = 512.0f * sdw2;
        ws[2049] = 1024.0f * sdwb;
        ws[2050] = 512.0f * sdb2 + EPSV;
    }
}

// ---------------- Stage 2: WMMA rank-3 expansion + NT stores ----------------
// One wave: 16 scalars x 512 channels = 32 k-tiles of 16x16.
// A (16x4 f32): lanes 0-15: {x*inv, inv}; lanes 16-31 (K2,K3): {1, 0}
// B ( 4x16 f32): lanes 0-15: {P[n], Q[n]}; lanes 16-31: {beta[n], 0}
//   -> one ds_load_b64 per lane per tile, no predication, no cndmask
// C = 0 (loop-invariant zero vector)
// D (16x16 f32): VGPR r, lanes 0-15 -> (M=r, N=lane); lanes 16-31 -> (M=r+8, N=lane-16)

__global__ void __launch_bounds__(256)
ln_wmma_kernel(const float* __restrict__ x,
               const float* __restrict__ ws,
               float* __restrict__ out,
               int n_scalars) {
    __shared__ v2f sComb[1024];  // [0,512): {P,Q}  [512,1024): {beta,0}

    const v2f* __restrict__ wsv = (const v2f*)ws;
    for (int i = threadIdx.x; i < 1024; i += blockDim.x) sComb[i] = wsv[i];
    const float a_c  = ws[2048];
    const float b2_c = ws[2049];
    const float c_c  = ws[2050];
    __syncthreads();

    const int lane     = threadIdx.x & 31;
    const int wave_gid = (int)(blockIdx.x * (blockDim.x >> 5)) + (threadIdx.x >> 5);
    const int base     = wave_gid * 16;   // first scalar of this wave's group
    if (base >= n_scalars) return;        // wave-uniform: EXEC stays all-1s at WMMA

    const int  m  = lane & 15;
    const bool lo = lane < 16;

    const float xv  = x[base + m];        // hi lanes load duplicates (harmless)
    const float inv = rsqrtf(fmaf(xv, fmaf(xv, a_c, b2_c), c_c));

    v2f A;
    A.x = lo ? xv * inv : 1.0f;           // K0 col | K2 col (ones -> +beta)
    A.y = lo ? inv      : 0.0f;           // K1 col | K3 col (zero)

    const v8f C = {0.f, 0.f, 0.f, 0.f, 0.f, 0.f, 0.f, 0.f};

    const int b_off = lo ? 0 : 512;       // hi half reads the {beta,0} rows
    // lanes 0-15 write rows 0..7 (VGPR r -> M=r); lanes 16-31 rows 8..15
    float* __restrict__ out_lane =
        out + (size_t)base * SIZE + (size_t)(lo ? 0 : 8) * SIZE + m;

#pragma unroll 4
    for (int t = 0; t < SIZE / 16; ++t) {
        const v2f B = sComb[b_off + t * 16 + m];   // single ds_load_b64, all lanes

        const v8f D = __builtin_amdgcn_wmma_f32_16x16x4_f32(
            /*neg_a=*/false, A, /*neg_b=*/false, B,
            /*c_mod=*/(short)0, C, /*reuse_a=*/false, /*reuse_b=*/false);

        float* __restrict__ p = out_lane + t * 16;
#pragma unroll
        for (int r = 0; r < 8; ++r)
            __builtin_nontemporal_store(D[r], p + (size_t)r * SIZE);
    }
}

// ---------------------------------------------------------------------------

extern "C" void kernel_launch(void* const* d_in, const int* in_sizes, int n_in,
                              void* d_out, int out_size, void* d_ws, size_t ws_size,
                              hipStream_t stream) {
    const float* x     = (const float*)d_in[0];  // (8192, 32)
    const float* w     = (const float*)d_in[1];  // (512,)
    const float* b     = (const float*)d_in[2];  // (512,)
    const float* gamma = (const float*)d_in[3];  // (512,)
    const float* beta  = (const float*)d_in[4];  // (512,)
    float*       out   = (float*)d_out;          // (8192, 32, 512)
    float*       ws    = (float*)d_ws;           // needs 2051*4 B ~= 8.2 KB

    // Stage 1: channel-coefficient precompute (one 512-thread block)
    ln_stats_kernel<<<1, SIZE, 0, stream>>>(w, b, gamma, beta, ws);

    // Stage 2: 1 wave per 16 scalars, 8 waves (256 threads) per block
    const int n_scalars = in_sizes[0];                 // 262144
    const int waves     = (n_scalars + 15) / 16;       // 16384
    const int blocks    = (waves + 7) / 8;             // 2048
    ln_wmma_kernel<<<blocks, 256, 0, stream>>>(x, ws, out, n_scalars);
}